// AELoss_37632503447761
// MI455X (gfx1250) — compile-verified
//
#include <hip/hip_runtime.h>
#include <hip/hip_bf16.h>

typedef __attribute__((ext_vector_type(2))) float v2f;
typedef __attribute__((ext_vector_type(8))) float v8f;

#define AE_M 30   // max persons
#define AE_K 17   // joints per person

__device__ __forceinline__ float wave_sum(float x) {
#pragma unroll
  for (int off = 16; off > 0; off >>= 1)
    x += __shfl_xor(x, off, 32);
  return x;
}

// One wave (32 lanes) per batch image. Lane m = person m (lanes 30,31 padded).
__global__ __launch_bounds__(32) void ae_loss_kernel(
    const float* __restrict__ tags, const int* __restrict__ joints,
    float* __restrict__ out, int B, long long N) {
  const int b = blockIdx.x;
  const int lane = threadIdx.x;

  __shared__ float s_mean[32];
  __shared__ float s_sq[32];
  __shared__ float s_val[32];

  float g[AE_K];
  float vz[AE_K];
  float sum = 0.f, cnt = 0.f;

  if (lane < AE_M) {
    const long long jbase = ((long long)b * AE_M + lane) * AE_K * 2;
    const long long tbase = (long long)b * N;
#pragma unroll
    for (int k = 0; k < AE_K; ++k) {
      int idx  = joints[jbase + 2 * k];
      int visr = joints[jbase + 2 * k + 1];
      float v = (visr > 0) ? 1.f : 0.f;
      float t = tags[tbase + (long long)idx];  // scattered gather (510/batch)
      g[k] = t;
      vz[k] = v;
      sum += t * v;
      cnt += v;
    }
  } else {
#pragma unroll
    for (int k = 0; k < AE_K; ++k) { g[k] = 0.f; vz[k] = 0.f; }
  }

  const float safe = fmaxf(cnt, 1.f);
  const float mean = sum / safe;
  float pp = 0.f;
#pragma unroll
  for (int k = 0; k < AE_K; ++k) {
    float d = g[k] - mean;
    pp += d * d * vz[k];
  }
  pp /= safe;
  const float valid = (cnt > 0.f) ? 1.f : 0.f;
  const float m_eff = mean * valid;  // invalid persons contribute mean 0 (masked later)

  s_mean[lane] = m_eff;
  s_sq[lane]   = m_eff * m_eff;
  s_val[lane]  = valid;
  __syncthreads();

  const float nval     = wave_sum(valid);
  const float pull_sum = wave_sum(pp * valid);

  // ---- push: 32x32 outer product mean*mean^T via 4x V_WMMA_F32_16X16X4_F32 ----
  // exp(-(mi-mj)^2) = exp(2*mi*mj - mi^2 - mj^2); D tile supplies mi*mj.
  float push_acc = 0.f;
#pragma unroll
  for (int r = 0; r < 2; ++r) {
    v2f a;  // A 16x4: only K=0 column nonzero -> VGPR0 lanes 0-15
    a.x = (lane < 16) ? s_mean[16 * r + (lane & 15)] : 0.f;
    a.y = 0.f;
#pragma unroll
    for (int c = 0; c < 2; ++c) {
      v2f bm;  // B 4x16: only K=0 row nonzero -> VGPR0 lanes 0-15
      bm.x = (lane < 16) ? s_mean[16 * c + (lane & 15)] : 0.f;
      bm.y = 0.f;
      v8f acc = {};
      acc = __builtin_amdgcn_wmma_f32_16x16x4_f32(
          /*neg_a=*/false, a, /*neg_b=*/false, bm,
          /*c_mod=*/(short)0, acc, /*reuse_a=*/false, /*reuse_b=*/false);
      const int   j   = 16 * c + (lane & 15);
      const float sqj = s_sq[j];
      const float vj  = s_val[j];
      const int rbase = 16 * r + ((lane >> 4) << 3);  // D: VGPR v -> row rbase+v
#pragma unroll
      for (int v = 0; v < 8; ++v) {
        const int i = rbase + v;
        push_acc += s_val[i] * vj * __expf(2.f * acc[v] - s_sq[i] - sqj);
      }
    }
  }
  const float push_total = wave_sum(push_acc);

  if (lane == 0) {
    const float n = nval;
    const float pull = (n > 0.f) ? (pull_sum / fmaxf(n, 1.f)) : 0.f;
    const float push =
        (n >= 2.f) ? ((push_total - n) / fmaxf((n - 1.f) * n, 1.f)) * 0.5f : 0.f;
    out[b]     = push;  // push first (reference returns (push, pull))
    out[B + b] = pull;
  }
}

extern "C" void kernel_launch(void* const* d_in, const int* in_sizes, int n_in,
                              void* d_out, int out_size, void* d_ws, size_t ws_size,
                              hipStream_t stream) {
  const float* tags   = (const float*)d_in[0];
  const int*   joints = (const int*)d_in[1];
  float*       out    = (float*)d_out;

  const int B = in_sizes[1] / (AE_M * AE_K * 2);  // joints is [B,30,17,2]
  const long long N = (long long)in_sizes[0] / (B > 0 ? B : 1);

  ae_loss_kernel<<<B, 32, 0, stream>>>(tags, joints, out, B, N);
}